// Affinity1d_35880156791045
// MI455X (gfx1250) — compile-verified
//
#include <hip/hip_runtime.h>

// Problem constants from the reference: B, H, NIN, C, E = (8, 1024, 256, 128, 16)
#define B_   8
#define H_   1024
#define NIN_ 256
#define C_   128
#define E_   16

typedef __attribute__((ext_vector_type(2))) float v2f;
typedef __attribute__((ext_vector_type(4))) float f32x4;
typedef __attribute__((ext_vector_type(8))) float v8f;

// ---------------------------------------------------------------------------
// Kernel A: s_t[b,h] = sum_c w_t[c] * (Wt @ x[b])[c,h]   (and same for phi)
// Implemented as a real fp32 WMMA GEMM (v_wmma_f32_16x16x4_f32), with the
// w-vector reduction fused on the D tiles in registers.
//
// One wave (32 lanes) owns a 16-wide h-strip of one batch. It keeps all 8
// c-tiles (C=128) as 8 v8f accumulators, loops K=NIN in steps of 4, loading
// the x (B-operand) tile once per k-step and reusing it across the 8 A tiles.
//
// WMMA f32 16x16x4 operand layouts (ISA 7.12.2):
//   A 16x4 : lane L -> M = L%16 ; VGPR j -> K = (L/16)*2 + j
//   B 4x16 : lane L -> N = L%16 ; VGPR j -> K = (L/16)*2 + j
//   C/D    : lane L -> N = L%16 ; VGPR r -> M = r + (L/16)*8
// ---------------------------------------------------------------------------
__global__ __launch_bounds__(128) void affinity_spr_wmma(
    const float* __restrict__ x,    // (B, NIN, H)
    const float* __restrict__ Wt,   // (C, NIN)
    const float* __restrict__ Wp,   // (C, NIN)
    const float* __restrict__ Wcat, // (E + 2C,)
    float* __restrict__ s_t,        // (B, H)
    float* __restrict__ s_p)        // (B, H)
{
    const int lane = threadIdx.x & 31;
    const int wave = threadIdx.x >> 5;
    const int b    = blockIdx.y;
    const int hb   = (blockIdx.x * 4 + wave) * 16;   // 16-column h strip
    const int half = lane >> 4;                      // 0 or 1 (upper lane half)
    const int l16  = lane & 15;
    const int koff = half * 2;

    for (int phase = 0; phase < 2; ++phase) {
        const float* __restrict__ W  = (phase == 0) ? Wt : Wp;
        const float* __restrict__ wv = Wcat + E_ + phase * C_;

        v8f acc[8] = {};   // 8 c-tiles x (16x16 f32) accumulators

        for (int kt = 0; kt < NIN_; kt += 4) {
            // B operand: x[b, kt+koff+j, hb+l16], j = 0,1 (stride H over K)
            const float* xp = x + ((size_t)b * NIN_ + (size_t)(kt + koff)) * H_
                                + hb + l16;
            v2f bm;
            bm.x = xp[0];
            bm.y = xp[H_];

#pragma unroll
            for (int ct = 0; ct < 8; ++ct) {
                // A operand: W[ct*16 + l16, kt+koff+j]
                const float* ap = W + (size_t)(ct * 16 + l16) * NIN_ + kt + koff;
                v2f am;
                am.x = ap[0];
                am.y = ap[1];
                acc[ct] = __builtin_amdgcn_wmma_f32_16x16x4_f32(
                    /*neg_a=*/false, am, /*neg_b=*/false, bm,
                    /*c_mod=*/(short)0, acc[ct],
                    /*reuse_a=*/false, /*reuse_b=*/false);
            }
        }

        // Fold w-vector over C in-register: lane L holds column h = hb + l16,
        // rows m = r + half*8 of each c-tile.
        float partial = 0.f;
#pragma unroll
        for (int ct = 0; ct < 8; ++ct)
#pragma unroll
            for (int r = 0; r < 8; ++r)
                partial += wv[ct * 16 + half * 8 + r] * acc[ct][r];

        // Combine the two lane halves (same h column, disjoint row sets).
        partial += __shfl_xor(partial, 16, 32);

        if (half == 0) {
            float* s = (phase == 0) ? s_t : s_p;
            s[b * H_ + hb + l16] = partial;
        }
    }
}

// ---------------------------------------------------------------------------
// Kernel B: pure-bandwidth stream.
// out[b,h,w] = sum_e w_e[e]*edges[b,e,h,w] + s_t[b,h] + s_p[b,w]
// One thread = 4 consecutive w (float4). 16 non-temporal b128 loads in
// flight per thread (edges are read-once: keep them out of the caches).
// ---------------------------------------------------------------------------
__global__ __launch_bounds__(256) void affinity_eterm(
    const float* __restrict__ edges, // (B, E, H, H)
    const float* __restrict__ Wcat,
    const float* __restrict__ s_t,   // (B, H)
    const float* __restrict__ s_p,   // (B, H)
    float* __restrict__ out)         // (B, 1, H, H)
{
    const unsigned idx4 = blockIdx.x * 256u + threadIdx.x; // unit = float4
    const unsigned b    = idx4 >> 18;          // / (H*H/4) = / 262144
    const unsigned rem  = idx4 & 0x3FFFFu;
    const unsigned h    = rem >> 8;            // / (H/4) = / 256
    const unsigned w4   = (rem & 255u) << 2;   // starting w

    float we[E_];
#pragma unroll
    for (int e = 0; e < E_; ++e) we[e] = Wcat[e];   // uniform -> scalar loads

    const size_t HW   = (size_t)H_ * H_;
    const float* base = edges + (size_t)b * E_ * HW + (size_t)h * H_ + w4;

    f32x4 acc = {};
#pragma unroll
    for (int e = 0; e < E_; ++e) {
        f32x4 v = __builtin_nontemporal_load(
            (const f32x4*)(base + (size_t)e * HW));
        acc += v * we[e];
    }

    const f32x4 sp = *(const f32x4*)(s_p + b * H_ + w4);
    const float st = s_t[b * H_ + h];
    acc = acc + sp + st;

    __builtin_nontemporal_store(acc, (f32x4*)(out + (size_t)idx4 * 4));
}

// ---------------------------------------------------------------------------
// Launch. Input order: adj(0, UNUSED by reference), edges(1), x(2), Wt(3),
// Wp(4), Wcat(5). Output: (B,1,H,H) fp32. Workspace: 64 KB for s_t/s_p.
// ---------------------------------------------------------------------------
extern "C" void kernel_launch(void* const* d_in, const int* in_sizes, int n_in,
                              void* d_out, int out_size, void* d_ws, size_t ws_size,
                              hipStream_t stream) {
    (void)in_sizes; (void)n_in; (void)out_size; (void)ws_size;

    const float* edges = (const float*)d_in[1];
    const float* x     = (const float*)d_in[2];
    const float* Wt    = (const float*)d_in[3];
    const float* Wp    = (const float*)d_in[4];
    const float* Wcat  = (const float*)d_in[5];
    float* out = (float*)d_out;

    float* s_t = (float*)d_ws;          // B*H floats
    float* s_p = s_t + B_ * H_;         // B*H floats

    // Kernel A: 512 waves total, 4 waves/block -> grid (16, 8)
    dim3 gridA(H_ / 16 / 4, B_);
    affinity_spr_wmma<<<gridA, 128, 0, stream>>>(x, Wt, Wp, Wcat, s_t, s_p);

    // Kernel B: B*H*H/4 float4 units, 256 threads/block -> 8192 blocks
    const unsigned total4 = (unsigned)((size_t)B_ * H_ * H_ / 4);
    affinity_eterm<<<total4 / 256, 256, 0, stream>>>(edges, Wcat, s_t, s_p, out);
}